// MultiGraphClassifier_32375463477758
// MI455X (gfx1250) — compile-verified
//
#include <hip/hip_runtime.h>
#include <math.h>

typedef __attribute__((ext_vector_type(2))) float v2f;
typedef __attribute__((ext_vector_type(8))) float v8f;

#define HIDN 128

// ---------------- degree kernels ----------------
__global__ void mgc_count_deg(const int* __restrict__ src, const int* __restrict__ dst,
                              float* __restrict__ degOut, float* __restrict__ degIn, int E) {
  int t = blockIdx.x * blockDim.x + threadIdx.x;
  if (t < E) {
    atomicAdd(&degOut[src[t]], 1.0f);
    atomicAdd(&degIn[dst[t]], 1.0f);
  }
}

__global__ void mgc_inv_sqrt_deg(float* __restrict__ degOut, float* __restrict__ degIn, int N) {
  int t = blockIdx.x * blockDim.x + threadIdx.x;
  if (t < N) {
    degOut[t] = rsqrtf(fmaxf(degOut[t], 1.0f));
    degIn[t]  = rsqrtf(fmaxf(degIn[t], 1.0f));
  }
}

// ---------------- WMMA fp32 GEMM: C[N x 128] = (A[N x K] * sOut[row]) @ W[K x 128] ----
// Block = 256 threads = 8 waves. blockIdx.x = row tile (16 rows). wave id = col tile.
// V_WMMA_F32_16X16X4_F32 layouts (ISA 7.12.2):
//   A 16x4: lanes 0-15 -> M=lane, V0=K0,V1=K1 ; lanes 16-31 -> M=lane-16, V0=K2,V1=K3
//   B 4x16: lanes 0-15 -> N=lane, V0=K0,V1=K1 ; lanes 16-31 -> N=lane-16, V0=K2,V1=K3
//   C/D:    lanes 0-15 -> N=lane, VGPR r = M=r ; lanes 16-31 -> M=r+8
__global__ void mgc_gemm_scaled_wmma(const float* __restrict__ A,
                                     const float* __restrict__ W,
                                     const float* __restrict__ sOut,
                                     float* __restrict__ C,
                                     int N, int K) {
  const int wave = threadIdx.x >> 5;
  const int lane = threadIdx.x & 31;
  const int row0 = blockIdx.x * 16;
  const int col0 = wave * 16;
  const int m    = lane & 15;          // M for A-frag, N for B-frag, N for C-frag
  const int kh   = (lane >> 4) * 2;    // 0 or 2: K sub-offset for this half-wave

  const float scale = sOut[row0 + m];
  const float* Arow = A + (size_t)(row0 + m) * K;

  v8f acc = {};
  for (int k0 = 0; k0 < K; k0 += 4) {
    v2f a, b;
    a.x = Arow[k0 + kh]     * scale;
    a.y = Arow[k0 + kh + 1] * scale;
    b.x = W[(size_t)(k0 + kh)     * HIDN + col0 + m];
    b.y = W[(size_t)(k0 + kh + 1) * HIDN + col0 + m];
    acc = __builtin_amdgcn_wmma_f32_16x16x4_f32(
        /*neg_a=*/false, a, /*neg_b=*/false, b,
        /*c_mod=*/(short)0, acc, /*reuse_a=*/false, /*reuse_b=*/false);
  }

  const int rbase = row0 + (lane >> 4) * 8;
#pragma unroll
  for (int r = 0; r < 8; ++r) {
    C[(size_t)(rbase + r) * HIDN + col0 + m] = acc[r];
  }
}

// ---------------- edge scatter: agg[dst] += h[src], one wave per edge ----------------
__global__ void mgc_edge_scatter(const int* __restrict__ src, const int* __restrict__ dst,
                                 const float* __restrict__ h, float* __restrict__ agg, int E) {
  unsigned int t = blockIdx.x * blockDim.x + threadIdx.x;
  int e = (int)(t >> 5);
  if (e < E) {
    int c4 = (int)(t & 31u) * 4;
    const float4 v = *(const float4*)(h + (size_t)src[e] * HIDN + c4);
    float* ap = agg + (size_t)dst[e] * HIDN + c4;
    atomicAdd(ap + 0, v.x);
    atomicAdd(ap + 1, v.y);
    atomicAdd(ap + 2, v.z);
    atomicAdd(ap + 3, v.w);
  }
}

// ---------------- epilogue: out = relu(agg * sIn[row] + b[col]) ----------------
__global__ void mgc_node_epilogue(const float* __restrict__ agg, const float* __restrict__ sIn,
                                  const float* __restrict__ bias, float* __restrict__ out, int N) {
  int t = blockIdx.x * blockDim.x + threadIdx.x;
  if (t < N * HIDN) {
    int row = t >> 7;
    int col = t & 127;
    float v = agg[t] * sIn[row] + bias[col];
    out[t] = fmaxf(v, 0.0f);
  }
}

// ---------------- column reductions (block per column) ----------------
__global__ void mgc_col_mean(const float* __restrict__ x, float* __restrict__ emb, int N) {
  __shared__ float s[256];
  int col = blockIdx.x;
  float acc = 0.0f;
  for (int i = threadIdx.x; i < N; i += blockDim.x) acc += x[(size_t)i * HIDN + col];
  s[threadIdx.x] = acc;
  __syncthreads();
  for (int st = 128; st > 0; st >>= 1) {
    if ((int)threadIdx.x < st) s[threadIdx.x] += s[threadIdx.x + st];
    __syncthreads();
  }
  if (threadIdx.x == 0) emb[col] = s[0] / (float)N;
}

__global__ void mgc_col_max(const float* __restrict__ x, float* __restrict__ emb, int N) {
  __shared__ float s[256];
  int col = blockIdx.x;
  float acc = -INFINITY;
  for (int i = threadIdx.x; i < N; i += blockDim.x) acc = fmaxf(acc, x[(size_t)i * HIDN + col]);
  s[threadIdx.x] = acc;
  __syncthreads();
  for (int st = 128; st > 0; st >>= 1) {
    if ((int)threadIdx.x < st) s[threadIdx.x] = fmaxf(s[threadIdx.x], s[threadIdx.x + st]);
    __syncthreads();
  }
  if (threadIdx.x == 0) emb[col] = s[0];
}

// ---------------- finalize: norm01 both, sum, classify ----------------
__device__ __forceinline__ float mgc_bred(float v, float* s, int t, int op) {
  // op: 0=sum 1=min 2=max ; 128 threads
  s[t] = v;
  __syncthreads();
  for (int st = 64; st > 0; st >>= 1) {
    if (t < st) {
      float o = s[t + st];
      s[t] = (op == 0) ? (s[t] + o) : ((op == 1) ? fminf(s[t], o) : fmaxf(s[t], o));
    }
    __syncthreads();
  }
  float r = s[0];
  __syncthreads();
  return r;
}

__global__ void mgc_finalize(const float* __restrict__ embA, const float* __restrict__ embF,
                             const float* __restrict__ Wcls, const float* __restrict__ bcls,
                             float* __restrict__ out) {
  __shared__ float s[128];
  int t = threadIdx.x;  // 128 threads
  float ae = embA[t];
  float fe = embF[t];
  // norm01 is invariant to the standardization step: (x-min)/(max-min)
  float minA = mgc_bred(ae, s, t, 1);
  float maxA = mgc_bred(ae, s, t, 2);
  float minF = mgc_bred(fe, s, t, 1);
  float maxF = mgc_bred(fe, s, t, 2);
  float a = (ae - minA) / (maxA - minA);
  float f = (fe - minF) / (maxF - minF);
  // softmax over a length-1 axis == 1.0, so emb = a + f
  float e = a + f;
  for (int c = 0; c < 10; ++c) {
    float dot = mgc_bred(e * Wcls[c * HIDN + t], s, t, 0);
    if (t == 0) out[c] = dot + bcls[c];
    __syncthreads();
  }
}

// ---------------- host side ----------------
extern "C" void kernel_launch(void* const* d_in, const int* in_sizes, int n_in,
                              void* d_out, int out_size, void* d_ws, size_t ws_size,
                              hipStream_t stream) {
  (void)in_sizes; (void)n_in; (void)out_size; (void)ws_size;

  const int APIG_N = 50000, FCG_N = 100000;
  const int APIG_E = 800000, FCG_E = 1600000;
  const int APIG_D = 256, FCG_D = 128;

  const float* apig_feat = (const float*)d_in[0];
  const float* fcg_feat  = (const float*)d_in[1];
  const float* W_a1 = (const float*)d_in[2];
  const float* b_a1 = (const float*)d_in[3];
  const float* W_a2 = (const float*)d_in[4];
  const float* b_a2 = (const float*)d_in[5];
  const float* W_f1 = (const float*)d_in[6];
  const float* b_f1 = (const float*)d_in[7];
  const float* W_f2 = (const float*)d_in[8];
  const float* b_f2 = (const float*)d_in[9];
  // d_in[10] attn_w, d_in[11] attn_b: unused (softmax over length-1 axis == 1)
  const float* W_cls = (const float*)d_in[12];
  const float* b_cls = (const float*)d_in[13];
  const int* apig_src = (const int*)d_in[14];
  const int* apig_dst = (const int*)d_in[15];
  const int* fcg_src  = (const int*)d_in[16];
  const int* fcg_dst  = (const int*)d_in[17];

  float* out = (float*)d_out;

  // workspace layout
  float* bufH   = (float*)d_ws;                       // FCG_N * 128
  float* bufAgg = bufH   + (size_t)FCG_N * HIDN;      // FCG_N * 128
  float* bufX   = bufAgg + (size_t)FCG_N * HIDN;      // FCG_N * 128
  float* degOut = bufX   + (size_t)FCG_N * HIDN;      // FCG_N
  float* degIn  = degOut + FCG_N;                     // FCG_N
  float* embA   = degIn  + FCG_N;                     // 128
  float* embF   = embA   + HIDN;                      // 128

  const int TB = 256;

  // ================= APIG graph =================
  hipMemsetAsync(degOut, 0, (size_t)2 * FCG_N * sizeof(float), stream);
  mgc_count_deg<<<(APIG_E + TB - 1) / TB, TB, 0, stream>>>(apig_src, apig_dst, degOut, degIn, APIG_E);
  mgc_inv_sqrt_deg<<<(APIG_N + TB - 1) / TB, TB, 0, stream>>>(degOut, degIn, APIG_N);

  // layer 1
  mgc_gemm_scaled_wmma<<<APIG_N / 16, TB, 0, stream>>>(apig_feat, W_a1, degOut, bufH, APIG_N, APIG_D);
  hipMemsetAsync(bufAgg, 0, (size_t)APIG_N * HIDN * sizeof(float), stream);
  mgc_edge_scatter<<<((size_t)APIG_E * 32 + TB - 1) / TB, TB, 0, stream>>>(apig_src, apig_dst, bufH, bufAgg, APIG_E);
  mgc_node_epilogue<<<((size_t)APIG_N * HIDN + TB - 1) / TB, TB, 0, stream>>>(bufAgg, degIn, b_a1, bufX, APIG_N);

  // layer 2
  mgc_gemm_scaled_wmma<<<APIG_N / 16, TB, 0, stream>>>(bufX, W_a2, degOut, bufH, APIG_N, HIDN);
  hipMemsetAsync(bufAgg, 0, (size_t)APIG_N * HIDN * sizeof(float), stream);
  mgc_edge_scatter<<<((size_t)APIG_E * 32 + TB - 1) / TB, TB, 0, stream>>>(apig_src, apig_dst, bufH, bufAgg, APIG_E);
  mgc_node_epilogue<<<((size_t)APIG_N * HIDN + TB - 1) / TB, TB, 0, stream>>>(bufAgg, degIn, b_a2, bufX, APIG_N);

  mgc_col_mean<<<HIDN, TB, 0, stream>>>(bufX, embA, APIG_N);

  // ================= FCG graph =================
  hipMemsetAsync(degOut, 0, (size_t)2 * FCG_N * sizeof(float), stream);
  mgc_count_deg<<<(FCG_E + TB - 1) / TB, TB, 0, stream>>>(fcg_src, fcg_dst, degOut, degIn, FCG_E);
  mgc_inv_sqrt_deg<<<(FCG_N + TB - 1) / TB, TB, 0, stream>>>(degOut, degIn, FCG_N);

  // layer 1
  mgc_gemm_scaled_wmma<<<FCG_N / 16, TB, 0, stream>>>(fcg_feat, W_f1, degOut, bufH, FCG_N, FCG_D);
  hipMemsetAsync(bufAgg, 0, (size_t)FCG_N * HIDN * sizeof(float), stream);
  mgc_edge_scatter<<<((size_t)FCG_E * 32 + TB - 1) / TB, TB, 0, stream>>>(fcg_src, fcg_dst, bufH, bufAgg, FCG_E);
  mgc_node_epilogue<<<((size_t)FCG_N * HIDN + TB - 1) / TB, TB, 0, stream>>>(bufAgg, degIn, b_f1, bufX, FCG_N);

  // layer 2
  mgc_gemm_scaled_wmma<<<FCG_N / 16, TB, 0, stream>>>(bufX, W_f2, degOut, bufH, FCG_N, HIDN);
  hipMemsetAsync(bufAgg, 0, (size_t)FCG_N * HIDN * sizeof(float), stream);
  mgc_edge_scatter<<<((size_t)FCG_E * 32 + TB - 1) / TB, TB, 0, stream>>>(fcg_src, fcg_dst, bufH, bufAgg, FCG_E);
  mgc_node_epilogue<<<((size_t)FCG_N * HIDN + TB - 1) / TB, TB, 0, stream>>>(bufAgg, degIn, b_f2, bufX, FCG_N);

  mgc_col_max<<<HIDN, TB, 0, stream>>>(bufX, embF, FCG_N);

  // ================= head =================
  mgc_finalize<<<1, HIDN, 0, stream>>>(embA, embF, W_cls, b_cls, out);
}